// SOHO_Pre_VD_20744692040154
// MI455X (gfx1250) — compile-verified
//
#include <hip/hip_runtime.h>
#include <hip/hip_bf16.h>
#include <float.h>
#include <math.h>

// CDNA5 / gfx1250: wave32, WMMA f32 16x16x4 (exact fp32 GEMM for ranking parity)
typedef __attribute__((ext_vector_type(2))) float v2f;
typedef __attribute__((ext_vector_type(8))) float v8f;

#define NROWS 32768
#define NTOK  8192
#define TDIM  768
#define BLK_ROWS 64     // 4 row-groups x 16 rows; x-tile = 64*772*4 = 193 KB LDS (dynamic)
#define XS_STRIDE 772   // 768 + 4 dword pad: A-fetch (b64 x32 lanes) hits all 64 LDS banks

// ---------------------------------------------------------------------------
// Kernel 1: row norms.  ws[0..32767] = ||x_i||^2 ; ws[32768..40959] = ||e_j||^2
// ---------------------------------------------------------------------------
__global__ __launch_bounds__(256) void vq_norms_kernel(const float* __restrict__ x,
                                                       const float* __restrict__ e,
                                                       float* __restrict__ ws) {
  const int wave = threadIdx.x >> 5;
  const int lane = threadIdx.x & 31;
  const int row  = blockIdx.x * 8 + wave;          // 0..40959
  const float* src = (row < NROWS) ? (x + (size_t)row * TDIM)
                                   : (e + (size_t)(row - NROWS) * TDIM);
  float acc = 0.f;
  for (int c = lane; c < TDIM; c += 32) { float v = src[c]; acc += v * v; }
  #pragma unroll
  for (int off = 16; off > 0; off >>= 1) acc += __shfl_down(acc, off, 32);
  if (lane == 0) ws[row] = acc;
}

// ---------------------------------------------------------------------------
// Kernel 2: fused  x@e^T -> s = ||e||^2 - 2*dot  -> per-row top-4 -> epilogues.
// Block = 256 thr (8 waves) owns 64 rows. Wave w = (row-group w>>1, col-half w&1):
// the 4 waves of a col-half fetch IDENTICAL embed tiles -> WGP$ serves 3/4 of B,
// cutting L2 traffic ~4x. Distance matrix (1 GB) is never materialized.
// ---------------------------------------------------------------------------
__global__ __launch_bounds__(256) void vq_topk_kernel(const float* __restrict__ x,
                                                      const float* __restrict__ embed,
                                                      const float* __restrict__ ws,
                                                      float* __restrict__ out) {
  extern __shared__ float smem[];                // 64*772 floats: x-tile, later cands

  const int tid  = threadIdx.x;
  const int wave = tid >> 5;
  const int lane = tid & 31;
  const int col  = lane & 15;      // A's M row selector and B's N column selector
  const int half = lane >> 4;      // K-offset selector (0 -> K+{0,1}, 1 -> K+{2,3})
  const int rg   = wave >> 1;      // row group 0..3
  const int ch   = wave & 1;       // column half 0..1
  const int row0 = blockIdx.x * BLK_ROWS;

  const float* enorm = ws + NROWS;

  // ---- stage x tile via CDNA5 async DMA (ASYNCcnt), no VGPR round-trip ----
  for (int q = tid; q < BLK_ROWS * (TDIM / 4); q += 256) {
    const int r = q / (TDIM / 4);
    const int c = (q - r * (TDIM / 4)) * 4;                 // 16B chunks
    const float* gp = x + (size_t)(row0 + r) * TDIM + c;
    const unsigned ldsoff = (unsigned)(size_t)(void*)(smem + r * XS_STRIDE + c);
    asm volatile("global_load_async_to_lds_b128 %0, %1, off"
                 :: "v"(ldsoff), "v"(gp) : "memory");
  }
  asm volatile("s_wait_asynccnt 0x0" ::: "memory");
  __syncthreads();

  // per-lane sorted top-4 (ascending) for the 8 rows this lane sees (row = m + 8*half)
  float tv0[8], tv1[8], tv2[8], tv3[8];
  int   ti0[8], ti1[8], ti2[8], ti3[8];
  #pragma unroll
  for (int m = 0; m < 8; ++m) {
    tv0[m] = tv1[m] = tv2[m] = tv3[m] = FLT_MAX;
    ti0[m] = ti1[m] = ti2[m] = ti3[m] = 0;
  }

  const float* aptr = smem + (rg * 16 + col) * XS_STRIDE + 2 * half;

  for (int t = 0; t < 256; ++t) {
    const int j = ch * 4096 + t * 16 + col;               // this lane's code id
    const float* bptr = embed + (size_t)j * TDIM + 2 * half;
    if (t + 1 < 256)                                      // warm next code tile
      __builtin_prefetch(embed + (size_t)(j + 16) * TDIM + 2 * half, 0, 0);
    const float en = enorm[j];

    v8f acc = {};                                         // C starts at inline 0
    #pragma unroll 8
    for (int k = 0; k < TDIM; k += 4) {
      v2f a = *(const v2f*)(aptr + k);    // ds_load_b64, bank-conflict-free
      v2f b = *(const v2f*)(bptr + k);    // global_load_b64, WGP$/L2-served
      acc = __builtin_amdgcn_wmma_f32_16x16x4_f32(
          /*neg_a=*/false, a, /*neg_b=*/false, b,
          /*c_mod=*/(short)0, acc, /*reuse_a=*/false, /*reuse_b=*/false);
    }

    // branchless bubble insertion: s = ||e||^2 - 2*dot (v_fma + v_min/v_max/v_cndmask)
    #pragma unroll
    for (int m = 0; m < 8; ++m) {
      float s  = fmaf(-2.0f, acc[m], en);
      int   sj = j;
      {
        const bool c0 = s < tv0[m];
        const float lo = fminf(s, tv0[m]); const float hi = fmaxf(s, tv0[m]);
        const int   li = c0 ? sj : ti0[m]; const int   hj = c0 ? ti0[m] : sj;
        tv0[m] = lo; ti0[m] = li; s = hi; sj = hj;
      }
      {
        const bool c1 = s < tv1[m];
        const float lo = fminf(s, tv1[m]); const float hi = fmaxf(s, tv1[m]);
        const int   li = c1 ? sj : ti1[m]; const int   hj = c1 ? ti1[m] : sj;
        tv1[m] = lo; ti1[m] = li; s = hi; sj = hj;
      }
      {
        const bool c2 = s < tv2[m];
        const float lo = fminf(s, tv2[m]); const float hi = fmaxf(s, tv2[m]);
        const int   li = c2 ? sj : ti2[m]; const int   hj = c2 ? ti2[m] : sj;
        tv2[m] = lo; ti2[m] = li; s = hi; sj = hj;
      }
      {
        const bool c3 = s < tv3[m];
        tv3[m] = fminf(s, tv3[m]);
        ti3[m] = c3 ? sj : ti3[m];
      }
    }
  }
  __syncthreads();   // all waves done with x-tile; reuse smem as candidate buffer

  float* cv   = smem;                  // [4 rg][2 ch][8 m][16 col][4 slot] = 4096 f
  int*   ci   = (int*)(smem + 4096);   // same shape, 4096 i
  int*   amin = (int*)(smem + 8192);   // 64 argmin indices

  const size_t QO = (size_t)NROWS * TDIM;     // output section offsets

  for (int round = 0; round < 2; ++round) {   // rows {..+0..7} then {..+8..15}
    if (half == round) {
      #pragma unroll
      for (int m = 0; m < 8; ++m) {
        const int base = ((((rg * 2 + ch) * 8 + m) * 16) + col) * 4;
        cv[base+0]=tv0[m]; cv[base+1]=tv1[m]; cv[base+2]=tv2[m]; cv[base+3]=tv3[m];
        ci[base+0]=ti0[m]; ci[base+1]=ti1[m]; ci[base+2]=ti2[m]; ci[base+3]=ti3[m];
      }
    }
    __syncthreads();

    if (tid < 32) {                           // thread q merges 128 candidates of a row
      const int q    = tid;
      const int mrg  = q >> 3;                // row group
      const int mrow = q & 7;
      const int rloc = mrg * 16 + round * 8 + mrow;   // row within block
      const int rgl  = row0 + rloc;                   // global row
      float m0=FLT_MAX,m1=FLT_MAX,m2=FLT_MAX,m3=FLT_MAX;
      int   x0=0,x1=0,x2=0,x3=0;
      for (int w = 0; w < 2; ++w)                     // col-half (ascending j order)
        for (int c = 0; c < 16; ++c) {
          const int base = ((((mrg * 2 + w) * 8 + mrow) * 16) + c) * 4;
          for (int s = 0; s < 4; ++s) {               // slots sorted ascending
            float v = cv[base + s];
            if (v >= m3) break;
            int vj = ci[base + s];
            {
              const bool b = v < m0;
              const float lo = fminf(v, m0); const float hi = fmaxf(v, m0);
              const int   li = b ? vj : x0;  const int   hj = b ? x0 : vj;
              m0 = lo; x0 = li; v = hi; vj = hj;
            }
            {
              const bool b = v < m1;
              const float lo = fminf(v, m1); const float hi = fmaxf(v, m1);
              const int   li = b ? vj : x1;  const int   hj = b ? x1 : vj;
              m1 = lo; x1 = li; v = hi; vj = hj;
            }
            {
              const bool b = v < m2;
              const float lo = fminf(v, m2); const float hi = fmaxf(v, m2);
              const int   li = b ? vj : x2;  const int   hj = b ? x2 : vj;
              m2 = lo; x2 = li; v = hi; vj = hj;
            }
            {
              const bool b = v < m3;
              m3 = fminf(v, m3);
              x3 = b ? vj : x3;
            }
          }
        }
      // restore true distances: d = s + ||x_row||^2 ; then reference epilogue
      const float xn = ws[rgl];
      const float d1 = m1 + xn, d2 = m2 + xn, d3 = m3 + xn;
      float nrm = sqrtf(d1*d1 + d2*d2 + d3*d3);
      nrm = fmaxf(nrm, 1e-12f);
      const float t1 = 1.0f / (d1 / nrm + 1e-4f);
      const float t2 = 1.0f / (d2 / nrm + 1e-4f);
      const float t3 = 1.0f / (d3 / nrm + 1e-4f);
      const float mx = fmaxf(t1, fmaxf(t2, t3));
      const float e1 = expf(t1 - mx), e2 = expf(t2 - mx), e3 = expf(t3 - mx);
      const float inv = 1.0f / (e1 + e2 + e3);

      out[QO + rgl] = (float)x0;                                  // encoding_indices
      float* tkv = out + QO + NROWS + (size_t)rgl * 3;            // topk_values
      tkv[0] = e1 * inv; tkv[1] = e2 * inv; tkv[2] = e3 * inv;
      float* tki = out + QO + NROWS + (size_t)NROWS * 3 + (size_t)rgl * 3;
      tki[0] = (float)x1; tki[1] = (float)x2; tki[2] = (float)x3; // topk_indices_out
      amin[rloc] = x0;
    }
    __syncthreads();
  }

  // quantize = embed[argmin]  (straight-through estimator is identity in value)
  for (int r = 0; r < BLK_ROWS; ++r) {
    const float4* erow = (const float4*)(embed + (size_t)amin[r] * TDIM);
    float4* orow = (float4*)(out + (size_t)(row0 + r) * TDIM);
    for (int c = tid; c < TDIM / 4; c += 256) orow[c] = erow[c];  // b128 copies
  }
}

// ---------------------------------------------------------------------------
extern "C" void kernel_launch(void* const* d_in, const int* in_sizes, int n_in,
                              void* d_out, int out_size, void* d_ws, size_t ws_size,
                              hipStream_t stream) {
  (void)in_sizes; (void)n_in; (void)out_size; (void)ws_size;
  const float* x  = (const float*)d_in[0];   // inputs_flatten [32768,768]
  const float* e  = (const float*)d_in[1];   // embed          [8192,768]
  float* out = (float*)d_out;
  float* ws  = (float*)d_ws;                 // 40960 floats of norms

  vq_norms_kernel<<<(NROWS + NTOK) / 8, 256, 0, stream>>>(x, e, ws);
  const size_t smem_bytes = (size_t)BLK_ROWS * XS_STRIDE * sizeof(float); // 193 KB
  vq_topk_kernel<<<NROWS / BLK_ROWS, 256, smem_bytes, stream>>>(x, e, ws, out);
}